// GCN_single_18348100289004
// MI455X (gfx1250) — compile-verified
//
#include <hip/hip_runtime.h>
#include <hip/hip_bf16.h>
#include <math.h>
#include <stdint.h>

typedef __attribute__((ext_vector_type(2))) float v2f;
typedef __attribute__((ext_vector_type(8))) float v8f;

#define NNODES 16384
#define NFEAT  256
#define NHID   64

// ---------------------------------------------------------------------------
// WMMA GEMM: C[M,64] = op(A[M,K] @ B[K,64] (+ bias) (relu))
// 256 threads = 8 waves; tile 128 rows x 64 cols; K chunk 32.
// A tiles staged by GLOBAL_LOAD_ASYNC_TO_LDS_B128 (double-buffered, DMA
// overlaps WMMA compute). B tile staged transposed so each WMMA fragment is
// one aligned ds_load_b64 (no repack movs). Stride 36: 16B-aligned rows,
// conflict-free (36*m mod 64 all distinct; upper half-lanes disjoint).
// ---------------------------------------------------------------------------
#define MT  128
#define KC  32
#define KCP 36

__device__ __forceinline__ void stage_A_async(
    const float* __restrict__ A, int m0, int kc, int K,
    float (*buf)[KCP], int tid)
{
    #pragma unroll
    for (int i = 0; i < 4; ++i) {
        int idx  = tid + i * 256;       // 0..1023 b128 transfers
        int row  = idx >> 3;            // 0..127
        int col4 = (idx & 7) << 2;      // 0,4,..,28
        unsigned long long ga =
            (unsigned long long)(A + (size_t)(m0 + row) * K + kc + col4);
        unsigned lds = (unsigned)(uintptr_t)&buf[row][col4];
        asm volatile("global_load_async_to_lds_b128 %0, %1, off"
                     :: "v"(lds), "v"(ga) : "memory");
    }
}

__global__ __launch_bounds__(256) void gemm64_wmma_kernel(
    const float* __restrict__ A, const float* __restrict__ B,
    const float* __restrict__ bias, float* __restrict__ C,
    int M, int K, int do_relu)
{
    __shared__ __align__(16) float sA[2][MT][KCP];
    __shared__ __align__(16) float sB[64][KCP];   // transposed: sB[n][k]

    const int tid   = threadIdx.x;
    const int wid   = tid >> 5;
    const int lane  = tid & 31;
    const int mloc  = lane & 15;
    const int kpair = (lane >> 4) << 1;   // 0 or 2
    const int rtop  = lane >> 4;
    const int m0    = blockIdx.x * MT;
    const int mbase = wid * 16;

    v8f acc[4] = {v8f{}, v8f{}, v8f{}, v8f{}};

    stage_A_async(A, m0, 0, K, sA[0], tid);
    int cur = 0;

    for (int kc = 0; kc < K; kc += KC) {
        // ---- stage B tile (32x64) transposed into sB[n][k] ----
        #pragma unroll
        for (int i = 0; i < 2; ++i) {
            int idx  = tid + i * 256;   // 0..511
            int row  = idx >> 4;        // k: 0..31
            int col4 = (idx & 15) << 2; // n: 0..60
            float4 v = *(const float4*)(B + (size_t)(kc + row) * 64 + col4);
            sB[col4 + 0][row] = v.x;
            sB[col4 + 1][row] = v.y;
            sB[col4 + 2][row] = v.z;
            sB[col4 + 3][row] = v.w;
        }
        asm volatile("s_wait_asynccnt 0x0" ::: "memory");
        __syncthreads();

        // DMA next A chunk into the other buffer while we compute
        if (kc + KC < K)
            stage_A_async(A, m0, kc + KC, K, sA[cur ^ 1], tid);

        const float* sArow = &sA[cur][mbase + mloc][0];
        #pragma unroll
        for (int ks = 0; ks < KC; ks += 4) {
            v2f a = *(const v2f*)(sArow + ks + kpair);
            #pragma unroll
            for (int nt = 0; nt < 4; ++nt) {
                v2f b = *(const v2f*)(&sB[nt * 16 + mloc][ks + kpair]);
                acc[nt] = __builtin_amdgcn_wmma_f32_16x16x4_f32(
                    false, a, false, b, (short)0, acc[nt], false, false);
            }
        }
        __syncthreads();
        cur ^= 1;
    }

    // ---- epilogue: VGPR i -> M = i + 8*rtop, N = mloc within 16-tile ----
    #pragma unroll
    for (int nt = 0; nt < 4; ++nt) {
        int col = nt * 16 + mloc;
        float bv = bias ? bias[col] : 0.0f;
        #pragma unroll
        for (int i = 0; i < 8; ++i) {
            int row = m0 + mbase + i + 8 * rtop;
            float v = acc[nt][i] + bv;
            if (do_relu) v = fmaxf(v, 0.0f);
            C[(size_t)row * 64 + col] = v;
        }
    }
}

// ---------------------------------------------------------------------------
// t2[row][0:2] = h[row][:] @ W2   (h: [N,64], W2: [64,2])
// ---------------------------------------------------------------------------
__global__ __launch_bounds__(256) void rowdot2_kernel(
    const float* __restrict__ h, const float* __restrict__ W2,
    float* __restrict__ t2)
{
    int row = blockIdx.x * blockDim.x + threadIdx.x;
    const float* hr = h + (size_t)row * NHID;
    float a0 = 0.0f, a1 = 0.0f;
    #pragma unroll 8
    for (int j = 0; j < NHID; ++j) {
        float v = hr[j];
        a0 += v * W2[j * 2 + 0];
        a1 += v * W2[j * 2 + 1];
    }
    t2[row * 2 + 0] = a0;
    t2[row * 2 + 1] = a1;
}

// ---------------------------------------------------------------------------
// y[row][0:2] = adj[row,:] @ t2 + b2   (one workgroup per row; pure BW)
// ---------------------------------------------------------------------------
__global__ __launch_bounds__(256) void adjvec2_kernel(
    const float* __restrict__ adj, const float* __restrict__ t2,
    const float* __restrict__ b2, float* __restrict__ y, int N)
{
    __shared__ float r0[256], r1[256];
    const int tid = threadIdx.x;
    const float* arow = adj + (size_t)blockIdx.x * N;
    float a0 = 0.0f, a1 = 0.0f;
    for (int k = tid * 4; k < N; k += 256 * 4) {
        float4 v = *(const float4*)(arow + k);
        a0 += v.x * t2[2*(k+0)] + v.y * t2[2*(k+1)]
            + v.z * t2[2*(k+2)] + v.w * t2[2*(k+3)];
        a1 += v.x * t2[2*(k+0)+1] + v.y * t2[2*(k+1)+1]
            + v.z * t2[2*(k+2)+1] + v.w * t2[2*(k+3)+1];
    }
    r0[tid] = a0; r1[tid] = a1;
    __syncthreads();
    for (int s = 128; s > 0; s >>= 1) {
        if (tid < s) { r0[tid] += r0[tid + s]; r1[tid] += r1[tid + s]; }
        __syncthreads();
    }
    if (tid == 0) {
        y[blockIdx.x * 2 + 0] = r0[0] + b2[0];
        y[blockIdx.x * 2 + 1] = r1[0] + b2[1];
    }
}

// ---------------------------------------------------------------------------
// out = max_rows(y) @ W3.T + b3     (single block)
// ---------------------------------------------------------------------------
__global__ __launch_bounds__(256) void maxpool_final_kernel(
    const float* __restrict__ y, const float* __restrict__ W3,
    const float* __restrict__ b3, float* __restrict__ out, int N)
{
    __shared__ float m0s[256], m1s[256];
    const int tid = threadIdx.x;
    float m0 = -3.402823466e38f, m1 = -3.402823466e38f;
    for (int i = tid; i < N; i += 256) {
        m0 = fmaxf(m0, y[i * 2 + 0]);
        m1 = fmaxf(m1, y[i * 2 + 1]);
    }
    m0s[tid] = m0; m1s[tid] = m1;
    __syncthreads();
    for (int s = 128; s > 0; s >>= 1) {
        if (tid < s) {
            m0s[tid] = fmaxf(m0s[tid], m0s[tid + s]);
            m1s[tid] = fmaxf(m1s[tid], m1s[tid + s]);
        }
        __syncthreads();
    }
    if (tid == 0)
        out[0] = m0s[0] * W3[0] + m1s[0] * W3[1] + b3[0];
}

// ---------------------------------------------------------------------------
extern "C" void kernel_launch(void* const* d_in, const int* in_sizes, int n_in,
                              void* d_out, int out_size, void* d_ws, size_t ws_size,
                              hipStream_t stream)
{
    (void)in_sizes; (void)n_in; (void)out_size; (void)ws_size;
    const float* x   = (const float*)d_in[0];   // [16384, 256]
    const float* adj = (const float*)d_in[1];   // [16384, 16384]
    const float* W1  = (const float*)d_in[2];   // [256, 64]
    const float* b1  = (const float*)d_in[3];   // [64]
    const float* W2  = (const float*)d_in[4];   // [64, 2]
    const float* b2  = (const float*)d_in[5];   // [2]
    const float* W3  = (const float*)d_in[6];   // [1, 2]
    const float* b3  = (const float*)d_in[7];   // [1]
    float* out = (float*)d_out;

    float* t1 = (float*)d_ws;                    // [16384, 64]
    float* h  = t1 + (size_t)NNODES * NHID;      // [16384, 64]
    float* t2 = h  + (size_t)NNODES * NHID;      // [16384, 2]
    float* y  = t2 + (size_t)NNODES * 2;         // [16384, 2]

    // 1) t1 = x @ W1                       (WMMA, no bias/relu)
    gemm64_wmma_kernel<<<NNODES / MT, 256, 0, stream>>>(
        x, W1, nullptr, t1, NNODES, NFEAT, 0);
    // 2) h = relu(adj @ t1 + b1)           (WMMA, dominant pass: adj stream #1)
    gemm64_wmma_kernel<<<NNODES / MT, 256, 0, stream>>>(
        adj, t1, b1, h, NNODES, NNODES, 1);
    // 3) t2 = h @ W2
    rowdot2_kernel<<<NNODES / 256, 256, 0, stream>>>(h, W2, t2);
    // 4) y = adj @ t2 + b2                 (adj stream #2, pure bandwidth)
    adjvec2_kernel<<<NNODES, 256, 0, stream>>>(adj, t2, b2, y, NNODES);
    // 5) out = max(y) @ W3.T + b3
    maxpool_final_kernel<<<1, 256, 0, stream>>>(y, W3, b3, out, NNODES);
}